// Butterfly_25220047962207
// MI455X (gfx1250) — compile-verified
//
#include <hip/hip_runtime.h>

typedef __attribute__((ext_vector_type(2))) float v2f;
typedef __attribute__((ext_vector_type(8))) float v8f;

#define IN_SIZE   1024
#define TILE_ROWS 16

// ws layout (floats):
//   [0,      16384) : W1T  64 blocks * 16*16 ; [b][j][i] = (S3 S2 S1 S0)_b[i][j]
//   [16384,  32768) : W2T  64 blocks * 16*16 ; blk=(G*16+c), [blk][j][i]
//   [32768,  36864) : W3   256 blocks * 4*4  ; [c][u_out][u_in] = (S9 S8)_c
#define W1T_OFF 0
#define W2T_OFF 16384
#define W3_OFF  32768

// twiddle flat: t[idx][q][i][j] at idx*2048 + q*4 + i*2 + j   (10,512,2,2)

__global__ void __launch_bounds__(256)
bfly_precompute(const float* __restrict__ tw, float* __restrict__ ws) {
    float* w1t = ws + W1T_OFF;
    float* w2t = ws + W2T_OFF;
    float* w3  = ws + W3_OFF;
    int tid = blockIdx.x * 256 + threadIdx.x;

    if (tid < 1024) {
        // ---- W1: stages 0..3, block b acts on indices [16b, 16b+16) ----
        int b = tid >> 4, j = tid & 15;
        float v[16];
#pragma unroll
        for (int l = 0; l < 16; ++l) v[l] = (l == j) ? 1.0f : 0.0f;
#pragma unroll
        for (int idx = 0; idx < 4; ++idx) {
            int s = 1 << idx;
            float nv[16];
#pragma unroll
            for (int l = 0; l < 16; ++l) {
                int p = b * 16 + l;
                int q = (p >> (idx + 1)) * s + (p & (s - 1));
                int i = (l >> idx) & 1;
                const float* tq = tw + idx * 2048 + q * 4 + i * 2;
                nv[l] = tq[0] * v[l & ~s] + tq[1] * v[l | s];
            }
#pragma unroll
            for (int l = 0; l < 16; ++l) v[l] = nv[l];
        }
#pragma unroll
        for (int i = 0; i < 16; ++i) w1t[b * 256 + j * 16 + i] = v[i];  // transposed
    } else if (tid < 2048) {
        // ---- W2: stages 4..7, block (G,c) acts on {256G + 16m + c} ----
        int t2 = tid - 1024;
        int blk = t2 >> 4, j = t2 & 15;
        int G = blk >> 4, c = blk & 15;
        float v[16];
#pragma unroll
        for (int m = 0; m < 16; ++m) v[m] = (m == j) ? 1.0f : 0.0f;
#pragma unroll
        for (int idx = 4; idx < 8; ++idx) {
            int s = 1 << idx;
            int bit = 1 << (idx - 4);
            float nv[16];
#pragma unroll
            for (int m = 0; m < 16; ++m) {
                int p = G * 256 + 16 * m + c;
                int q = (p >> (idx + 1)) * s + (p & (s - 1));
                int i = (p >> idx) & 1;
                const float* tq = tw + idx * 2048 + q * 4 + i * 2;
                nv[m] = tq[0] * v[m & ~bit] + tq[1] * v[m | bit];
            }
#pragma unroll
            for (int m = 0; m < 16; ++m) v[m] = nv[m];
        }
#pragma unroll
        for (int i = 0; i < 16; ++i) w2t[blk * 256 + j * 16 + i] = v[i];
    } else if (tid < 2304) {
        // ---- W3: stages 8..9, block c acts on {c + 256u} ----
        int c = tid - 2048;
        float v[4][4];
#pragma unroll
        for (int u = 0; u < 4; ++u)
#pragma unroll
            for (int j = 0; j < 4; ++j) v[u][j] = (u == j) ? 1.0f : 0.0f;
        float nv[4][4];
        // stage 8: stride 256, q = (u>>1)*256 + c, i = u&1
#pragma unroll
        for (int u = 0; u < 4; ++u) {
            const float* tq = tw + 8 * 2048 + ((u >> 1) * 256 + c) * 4 + (u & 1) * 2;
#pragma unroll
            for (int j = 0; j < 4; ++j)
                nv[u][j] = tq[0] * v[u & ~1][j] + tq[1] * v[u | 1][j];
        }
#pragma unroll
        for (int u = 0; u < 4; ++u)
#pragma unroll
            for (int j = 0; j < 4; ++j) v[u][j] = nv[u][j];
        // stage 9: stride 512, q = c + 256*(u&1), i = (u>>1)&1
#pragma unroll
        for (int u = 0; u < 4; ++u) {
            const float* tq = tw + 9 * 2048 + (c + 256 * (u & 1)) * 4 + ((u >> 1) & 1) * 2;
#pragma unroll
            for (int j = 0; j < 4; ++j)
                nv[u][j] = tq[0] * v[u & ~2][j] + tq[1] * v[u | 2][j];
        }
#pragma unroll
        for (int u = 0; u < 4; ++u)
#pragma unroll
            for (int j = 0; j < 4; ++j) w3[c * 16 + u * 4 + j] = nv[u][j];
    }
}

// XOR-swizzled LDS index: flips col bits [5:2] by row -> distinct banks when
// 16 lanes access the same column across the 16 rows (the dominant pattern).
__device__ __forceinline__ int SWZ(int r, int c) {
    return r * IN_SIZE + (c ^ ((r & 15) << 2));
}

__global__ void __launch_bounds__(256)
bfly_main(const float* __restrict__ x, const float* __restrict__ bias,
          const float* __restrict__ ws, float* __restrict__ out) {
    __shared__ float U[TILE_ROWS * IN_SIZE];   // 64 KB, in-place across stages

    const float* w1t = ws + W1T_OFF;
    const float* w2t = ws + W2T_OFF;
    const float* w3  = ws + W3_OFF;

    const int row0 = blockIdx.x * TILE_ROWS;
    const int tid  = threadIdx.x;
    const int wave = tid >> 5;
    const int lane = tid & 31;
    const int m16  = lane & 15;          // A: row M ; B: col N
    const int khi  = (lane >> 4) * 2;    // per-lane K base (ISA 16x4 layout)

    // ---- stage the 16x1024 fp32 tile into LDS (coalesced float4) ----
    {
        const float4* xg = (const float4*)(x + (size_t)row0 * IN_SIZE);
#pragma unroll
        for (int it = 0; it < 16; ++it) {
            int idx = it * 256 + tid;        // 0..4095 float4s
            int r   = idx >> 8;
            int c4  = idx & 255;
            float4 val = xg[r * 256 + c4];
            // swizzle at float4 granularity keeps 16B alignment
            *(float4*)&U[r * IN_SIZE + ((c4 * 4) ^ ((r & 15) << 2))] = val;
        }
    }
    __syncthreads();

    // ---- stage group 1: 64 x (16 rows)x(16x16 block), contiguous columns ----
    {
        v2f a[8][4];
#pragma unroll
        for (int t = 0; t < 8; ++t) {
            int b = wave + 8 * t;
#pragma unroll
            for (int k = 0; k < 4; ++k) {
                int col = b * 16 + k * 4 + khi;
                a[t][k].x = U[SWZ(m16, col)];
                a[t][k].y = U[SWZ(m16, col + 1)];
            }
        }
        __syncthreads();                      // all reads done before in-place writes
#pragma unroll
        for (int t = 0; t < 8; ++t) {
            int b = wave + 8 * t;
            v8f acc = {};
#pragma unroll
            for (int k = 0; k < 4; ++k) {
                int K = k * 4 + khi;
                v2f bf;
                bf.x = w1t[b * 256 + K * 16 + m16];
                bf.y = w1t[b * 256 + (K + 1) * 16 + m16];
                acc = __builtin_amdgcn_wmma_f32_16x16x4_f32(
                        false, a[t][k], false, bf, (short)0, acc, false, false);
            }
#pragma unroll
            for (int v = 0; v < 8; ++v) {
                int r = v + ((lane >> 4) << 3);
                U[SWZ(r, b * 16 + m16)] = acc[v];
            }
        }
    }
    __syncthreads();

    // ---- stage group 2: 64 blocks on {256G + 16m + c} (stride-16 gather) ----
    {
        v2f a[8][4];
#pragma unroll
        for (int t = 0; t < 8; ++t) {
            int blk = wave + 8 * t;
            int G = blk >> 4, c = blk & 15;
#pragma unroll
            for (int k = 0; k < 4; ++k) {
                int mb = k * 4 + khi;
                a[t][k].x = U[SWZ(m16, G * 256 + mb * 16 + c)];
                a[t][k].y = U[SWZ(m16, G * 256 + (mb + 1) * 16 + c)];
            }
        }
        __syncthreads();
#pragma unroll
        for (int t = 0; t < 8; ++t) {
            int blk = wave + 8 * t;
            int G = blk >> 4, c = blk & 15;
            v8f acc = {};
#pragma unroll
            for (int k = 0; k < 4; ++k) {
                int K = k * 4 + khi;
                v2f bf;
                bf.x = w2t[blk * 256 + K * 16 + m16];
                bf.y = w2t[blk * 256 + (K + 1) * 16 + m16];
                acc = __builtin_amdgcn_wmma_f32_16x16x4_f32(
                        false, a[t][k], false, bf, (short)0, acc, false, false);
            }
#pragma unroll
            for (int v = 0; v < 8; ++v) {
                int r = v + ((lane >> 4) << 3);
                U[SWZ(r, G * 256 + m16 * 16 + c)] = acc[v];
            }
        }
    }
    __syncthreads();

    // ---- stages 8-9 (4x4 across stride 256) + bias, coalesced store ----
    {
        int c = tid;                      // 0..255
        float m[16];
#pragma unroll
        for (int i = 0; i < 16; ++i) m[i] = w3[c * 16 + i];
        float bi[4];
#pragma unroll
        for (int u = 0; u < 4; ++u) bi[u] = bias[c + 256 * u];
#pragma unroll
        for (int r = 0; r < 16; ++r) {
            float in0 = U[SWZ(r, c)];
            float in1 = U[SWZ(r, c + 256)];
            float in2 = U[SWZ(r, c + 512)];
            float in3 = U[SWZ(r, c + 768)];
            float* og = out + (size_t)(row0 + r) * IN_SIZE;
#pragma unroll
            for (int u = 0; u < 4; ++u) {
                og[c + 256 * u] = m[u * 4 + 0] * in0 + m[u * 4 + 1] * in1 +
                                  m[u * 4 + 2] * in2 + m[u * 4 + 3] * in3 + bi[u];
            }
        }
    }
}

extern "C" void kernel_launch(void* const* d_in, const int* in_sizes, int n_in,
                              void* d_out, int out_size, void* d_ws, size_t ws_size,
                              hipStream_t stream) {
    const float* x    = (const float*)d_in[0];
    const float* tw   = (const float*)d_in[1];
    const float* bias = (const float*)d_in[2];
    float* out = (float*)d_out;
    float* ws  = (float*)d_ws;          // needs 36864 floats = 144 KB

    int rows = in_sizes[0] / IN_SIZE;   // 32768

    bfly_precompute<<<dim3(9), dim3(256), 0, stream>>>(tw, ws);
    bfly_main<<<dim3(rows / TILE_ROWS), dim3(256), 0, stream>>>(x, bias, ws, out);
}